// Roi_Extractor_27960237097553
// MI455X (gfx1250) — compile-verified
//
#include <hip/hip_runtime.h>
#include <stdint.h>

#define ROI 31
#define HALF 15
#define CCH 32
#define HH 256
#define WW 256
#define ROW_BYTES (ROI * CCH * 4)   // 3968 bytes per ROI row
#define BUF_FLOATS 1024             // row buffer padded to 4 KB
#define WAVES_PER_BLOCK 8

// Async DMA ops add INST_OFFSET to BOTH the LDS and the global address, so a
// single per-lane base pair + literal offsets covers the whole row.
#define A_LOAD(OFF) \
  asm volatile("global_load_async_to_lds_b128 %0, %1, off offset:" #OFF \
               :: "v"(laddr0), "v"(gload) : "memory")
#define A_STORE(OFF) \
  asm volatile("global_store_async_from_lds_b128 %0, %1, off offset:" #OFF \
               " th:TH_STORE_NT" \
               :: "v"(gstore), "v"(laddr0) : "memory")

__global__ __launch_bounds__(256)
void Roi_Extractor_27960237097553_kernel(const float* __restrict__ poses,
                                         const int*   __restrict__ centers,
                                         float*       __restrict__ out,
                                         int nroi)
{
    __shared__ __align__(16) float lds[WAVES_PER_BLOCK * BUF_FLOATS]; // 32 KB

    const int tid  = threadIdx.x;
    const int wave = tid >> 5;
    const int lane = tid & 31;

    const int r  = blockIdx.x * WAVES_PER_BLOCK + wave;  // ROI index
    const int dy = blockIdx.y;                           // row within ROI
    if (r >= nroi) return;

    const int b  = centers[r * 3 + 0];
    const int cy = centers[r * 3 + 1];
    const int cx = centers[r * 3 + 2];

    const int  row      = cy + dy - HALF;
    const bool rowValid = (row >= 0) && (row < HH);
    const int  xs       = cx - HALF;

    // Valid 16B-chunk range [lo8, hi8) within the 248-chunk row
    // (one pixel = 32 floats = 8 chunks; validity is pixel-granular).
    const int jlo = xs < 0 ? -xs : 0;
    const int jhi = (WW - xs) < ROI ? (WW - xs) : ROI;
    const int lo8 = rowValid ? jlo * 8 : 248;
    const int hi8 = rowValid ? jhi * 8 : 0;

    const int rclamp = rowValid ? row : 0;
    const uint64_t inRowBase = (uint64_t)poses +
        (uint64_t)(int64_t)(((int64_t)(b * HH + rclamp) * WW + (int64_t)xs) *
                            (CCH * 4));

    const unsigned ldsBase = (unsigned)(uintptr_t)(&lds[wave * BUF_FLOATS]);
    const unsigned laddr0  = ldsBase + (unsigned)lane * 16u;
    const uint64_t gload   = inRowBase + (uint64_t)lane * 16u;

    if (lo8 == 0 && hi8 == 248) {
        // Fully interior row (common case): straight async DMA global->LDS.
        A_LOAD(0);    A_LOAD(512);  A_LOAD(1024); A_LOAD(1536);
        A_LOAD(2048); A_LOAD(2560); A_LOAD(3072);
        if (lane < 24) A_LOAD(3584);
    } else {
        // Border row: zero the whole 4 KB buffer, then DMA the valid span.
        float4* zb = (float4*)&lds[wave * BUF_FLOATS];
        #pragma unroll
        for (int t = 0; t < 8; ++t)
            zb[t * 32 + lane] = make_float4(0.f, 0.f, 0.f, 0.f);
        asm volatile("s_wait_dscnt 0x0" ::: "memory");
        const int i0 = lane;
        if (i0       >= lo8 && i0       < hi8) A_LOAD(0);
        if (i0 + 32  >= lo8 && i0 + 32  < hi8) A_LOAD(512);
        if (i0 + 64  >= lo8 && i0 + 64  < hi8) A_LOAD(1024);
        if (i0 + 96  >= lo8 && i0 + 96  < hi8) A_LOAD(1536);
        if (i0 + 128 >= lo8 && i0 + 128 < hi8) A_LOAD(2048);
        if (i0 + 160 >= lo8 && i0 + 160 < hi8) A_LOAD(2560);
        if (i0 + 192 >= lo8 && i0 + 192 < hi8) A_LOAD(3072);
        if (i0 + 224 >= lo8 && i0 + 224 < hi8) A_LOAD(3584);
    }

    // All LDS writes (zeros + DMA) must land before the store engine reads.
    asm volatile("s_wait_asynccnt 0x0" ::: "memory");

    const uint64_t outRowBase =
        (uint64_t)out + (uint64_t)(r * ROI + dy) * (uint64_t)ROW_BYTES;
    const uint64_t gstore = outRowBase + (uint64_t)lane * 16u;

    // Stream the row LDS -> global (write-once data: non-temporal, keep the
    // L2 free for the 128 MB input which fits in the 192 MB L2).
    A_STORE(0);    A_STORE(512);  A_STORE(1024); A_STORE(1536);
    A_STORE(2048); A_STORE(2560); A_STORE(3072);
    if (lane < 24) A_STORE(3584);
    // S_ENDPGM's implicit wait-idle drains the outstanding async stores.
}

extern "C" void kernel_launch(void* const* d_in, const int* in_sizes, int n_in,
                              void* d_out, int out_size, void* d_ws, size_t ws_size,
                              hipStream_t stream) {
    const float* poses   = (const float*)d_in[0];
    const int*   centers = (const int*)d_in[1];
    float*       out     = (float*)d_out;

    const int nroi = in_sizes[1] / 3;                       // 4096
    dim3 grid((nroi + WAVES_PER_BLOCK - 1) / WAVES_PER_BLOCK, ROI);

    Roi_Extractor_27960237097553_kernel<<<grid, 256, 0, stream>>>(
        poses, centers, out, nroi);
}